// LAF_Extractor_29978871726647
// MI455X (gfx1250) — compile-verified
//
#include <hip/hip_runtime.h>
#include <hip/hip_bf16.h>

typedef __attribute__((ext_vector_type(2))) float v2f;
typedef __attribute__((ext_vector_type(8))) float v8f;

#define B_SZ    64
#define C_SF    256
#define H_SF    56
#define W_SF    56
#define NVERTS  1220
#define NSUB    305
#define TILE_N  16
#define NT      20          // ceil(305/16)
#define FEAT_LD 260         // padded strides (LDS has 64 banks; 260 % 64 != 0)
#define Y0_LD   132
#define Y1_LD   68

// D = A(16x4 f32) * B(4x16 f32) + C(16x16 f32), full-precision WMMA (CDNA5).
static __device__ inline v8f wmma4f32(v2f a, v2f b, v8f c) {
    // 8 args: (neg_a, A, neg_b, B, c_mod, C, reuse_a, reuse_b)
    return __builtin_amdgcn_wmma_f32_16x16x4_f32(false, a, false, b, (short)0, c,
                                                 false, false);
}

__global__ __launch_bounds__(256)
void laf_fused_kernel(const float* __restrict__ p,       // [B, NVERTS, 3]
                      const float* __restrict__ extr,    // [B, 4, 4]
                      const float* __restrict__ intr,    // [B, 4, 3]
                      const float* __restrict__ s_feat,  // [B, 256, 56, 56]
                      const int*   __restrict__ sub_idx, // [305]
                      const float* __restrict__ W0, const float* __restrict__ b0,
                      const float* __restrict__ W1, const float* __restrict__ b1,
                      const float* __restrict__ W2, const float* __restrict__ b2,
                      float* __restrict__ out_feat,      // [B, 5*305] (o-major)
                      float* __restrict__ out_proj)      // [B, 305, 2]
{
    __shared__ float featT[TILE_N][FEAT_LD];  // [n][c], c = 0..255
    __shared__ float y0T[TILE_N][Y0_LD];      // [n][m], m = 0..127
    __shared__ float y1T[TILE_N][Y1_LD];      // [n][m], m = 0..63
    __shared__ int   cOff[TILE_N][4];         // clamped y*56+x per corner
    __shared__ float cW[TILE_N][4];           // bilinear weight * validity

    const int blk  = blockIdx.x;
    const int b    = blk / NT;
    const int t    = blk % NT;
    const int tid  = threadIdx.x;
    const int lane = tid & 31;                // wave32
    const int wave = tid >> 5;

    //---------------- Stage A: projection + sampling setup (16 lanes) -------
    if (tid < TILE_N) {
        const int  n     = t * TILE_N + tid;
        const bool valid = (n < NSUB);
        const int  ni    = valid ? n : (NSUB - 1);
        const int  vidx  = sub_idx[ni];
        const float* pp  = p + ((size_t)b * NVERTS + vidx) * 3;
        const float px = pp[0], py = pp[1], pz = pp[2];

        const float* E = extr + b * 16;       // row-major [f][g]
        float cam[4];
#pragma unroll
        for (int g = 0; g < 4; ++g)
            cam[g] = px * E[0*4+g] + py * E[1*4+g] + pz * E[2*4+g] + E[3*4+g];

        const float* I = intr + b * 12;       // row-major [f][g]
        float img[3];
#pragma unroll
        for (int g = 0; g < 3; ++g)
            img[g] = cam[0]*I[0*3+g] + cam[1]*I[1*3+g] + cam[2]*I[2*3+g]
                   + cam[3]*I[3*3+g];

        const float invz = 1.0f / img[2];
        const float prx = img[0] * invz;
        const float pry = img[1] * invz;
        if (valid) {
            out_proj[((size_t)b * NSUB + n) * 2 + 0] = prx;
            out_proj[((size_t)b * NSUB + n) * 2 + 1] = pry;
        }

        // align_corners=True sample coord: x = p/224 * (W-1)
        const float sx = prx * ((float)(W_SF - 1) / 224.0f);
        const float sy = pry * ((float)(H_SF - 1) / 224.0f);
        const float fx0 = floorf(sx), fy0 = floorf(sy);
        const float wx1 = sx - fx0, wx0 = 1.0f - wx1;
        const float wy1 = sy - fy0, wy0 = 1.0f - wy1;
        const int ix0 = (int)fx0, iy0 = (int)fy0;
#pragma unroll
        for (int k = 0; k < 4; ++k) {
            const int  xi = ix0 + (k & 1);
            const int  yi = iy0 + (k >> 1);
            const bool v  = valid && xi >= 0 && xi < W_SF && yi >= 0 && yi < H_SF;
            const int  xc = min(max(xi, 0), W_SF - 1);
            const int  yc = min(max(yi, 0), H_SF - 1);
            const float w = ((k & 1) ? wx1 : wx0) * ((k >> 1) ? wy1 : wy0);
            cOff[tid][k] = yc * W_SF + xc;
            cW[tid][k]   = v ? w : 0.0f;
        }
    }
    __syncthreads();

    //---------------- Stage B: gather (thread = channel plane) --------------
    {
        // Per-(b,c) 12.5KB plane; all 20 n-tiles of batch b reuse it in L2.
        const float* plane = s_feat + ((size_t)b * C_SF + tid) * (H_SF * W_SF);
#pragma unroll
        for (int n = 0; n < TILE_N; ++n) {
            float acc = 0.0f;
#pragma unroll
            for (int k = 0; k < 4; ++k)
                acc += cW[n][k] * plane[cOff[n][k]];
            featT[n][tid] = acc;
        }
    }
    __syncthreads();

    const int ksub = (lane >> 4) * 2;   // 0 for lanes 0-15, 2 for lanes 16-31
    const int ncol = lane & 15;         // N index for B/C/D operands
    const int mhi  = (lane >= 16) ? 8 : 0;

    //---------------- Stage C0: layer0 = leakyrelu(W0[128,256] @ feat) ------
    {
        const int   mbase = wave * 16;
        const float* Wrow = W0 + (size_t)(mbase + (lane & 15)) * C_SF;
        v8f acc = {};
        for (int k0 = 0; k0 < C_SF; k0 += 4) {
            v2f a, bm;
            a.x  = Wrow[k0 + ksub];
            a.y  = Wrow[k0 + ksub + 1];
            bm.x = featT[ncol][k0 + ksub];
            bm.y = featT[ncol][k0 + ksub + 1];
            acc = wmma4f32(a, bm, acc);
        }
#pragma unroll
        for (int r = 0; r < 8; ++r) {
            const int   mo = mbase + r + mhi;
            float v = acc[r] + b0[mo];
            v = (v > 0.0f) ? v : 0.01f * v;
            y0T[ncol][mo] = v;
        }
    }
    __syncthreads();

    //---------------- Stage C1: layer1 = leakyrelu(W1[64,384] @ [y0;feat]) --
    if (wave < 4) {
        const int   mbase = wave * 16;
        const float* Wrow = W1 + (size_t)(mbase + (lane & 15)) * (128 + C_SF);
        v8f acc = {};
        for (int k0 = 0; k0 < 128 + C_SF; k0 += 4) {
            const int k = k0 + ksub;        // pair never straddles k=128
            v2f a, bm;
            a.x = Wrow[k];
            a.y = Wrow[k + 1];
            if (k < 128) { bm.x = y0T[ncol][k];         bm.y = y0T[ncol][k + 1]; }
            else         { bm.x = featT[ncol][k - 128]; bm.y = featT[ncol][k - 127]; }
            acc = wmma4f32(a, bm, acc);
        }
#pragma unroll
        for (int r = 0; r < 8; ++r) {
            const int   mo = mbase + r + mhi;
            float v = acc[r] + b1[mo];
            v = (v > 0.0f) ? v : 0.01f * v;
            y1T[ncol][mo] = v;
        }
    }
    __syncthreads();

    //---------------- Stage C2: layer2 = relu(W2[5,320] @ [y1;feat]) --------
    if (wave == 0) {
        const int   mrow   = lane & 15;
        const bool  mvalid = (mrow < 5);
        const float* Wrow  = W2 + (size_t)mrow * (64 + C_SF);
        v8f acc = {};
        for (int k0 = 0; k0 < 64 + C_SF; k0 += 4) {
            const int k = k0 + ksub;        // pair never straddles k=64
            v2f a, bm;
            a.x = mvalid ? Wrow[k]     : 0.0f;
            a.y = mvalid ? Wrow[k + 1] : 0.0f;
            if (k < 64) { bm.x = y1T[ncol][k];        bm.y = y1T[ncol][k + 1]; }
            else        { bm.x = featT[ncol][k - 64]; bm.y = featT[ncol][k - 63]; }
            acc = wmma4f32(a, bm, acc);
        }
        const int n_global = t * TILE_N + ncol;
#pragma unroll
        for (int r = 0; r < 8; ++r) {
            const int mo = r + mhi;
            if (mo < 5 && n_global < NSUB) {
                float v = acc[r] + b2[mo];
                v = (v > 0.0f) ? v : 0.0f;
                // y.reshape(B,-1): [b, o*305 + n]
                out_feat[(size_t)b * (5 * NSUB) + (size_t)mo * NSUB + n_global] = v;
            }
        }
    }
}

extern "C" void kernel_launch(void* const* d_in, const int* in_sizes, int n_in,
                              void* d_out, int out_size, void* d_ws, size_t ws_size,
                              hipStream_t stream) {
    const float* p      = (const float*)d_in[0];
    const float* extr   = (const float*)d_in[1];
    const float* intr   = (const float*)d_in[2];
    // d_in[3] = bbox: present in inputs but unused by the reference math.
    const float* s_feat = (const float*)d_in[4];
    const int*   s_idx  = (const int*)d_in[5];
    const float* W0     = (const float*)d_in[6];
    const float* b0     = (const float*)d_in[7];
    const float* W1     = (const float*)d_in[8];
    const float* b1     = (const float*)d_in[9];
    const float* W2     = (const float*)d_in[10];
    const float* b2     = (const float*)d_in[11];

    float* out      = (float*)d_out;
    float* out_feat = out;                               // 64*1525 floats
    float* out_proj = out + (size_t)B_SZ * 5 * NSUB;     // 64*305*2 floats

    dim3 grid(B_SZ * NT);   // 64 batches x 20 point-tiles = 1280 workgroups
    dim3 block(256);        // 8 wave32 waves
    hipLaunchKernelGGL(laf_fused_kernel, grid, block, 0, stream,
                       p, extr, intr, s_feat, s_idx,
                       W0, b0, W1, b1, W2, b2, out_feat, out_proj);
}